// MyModel_61289183314444
// MI455X (gfx1250) — compile-verified
//
#include <hip/hip_runtime.h>

typedef __attribute__((ext_vector_type(2))) float v2f;
typedef __attribute__((ext_vector_type(4))) float v4f;
typedef __attribute__((ext_vector_type(4))) int   v4i;
typedef __attribute__((ext_vector_type(8))) float v8f;

#define B_   8
#define S_   2048
#define H_   1024
#define NTOK (B_ * S_)

// ---------------- Phase 1: token projection via V_WMMA_F32_16X16X4_F32 -----
#define WAVES 4
#define TPW   16          // tokens per wave (WMMA M)
#define KC    64          // K chunk staged in LDS
#define AROW  68          // padded LDS row stride (floats): banks 4m+c, conflict-free
#define ATILE (TPW * AROW)

__global__ __launch_bounds__(128) void phase1_wmma(
    const float* __restrict__ rep, const int* __restrict__ mask,
    const float* __restrict__ Ws,  const float* __restrict__ bs,
    const float* __restrict__ We,  const float* __restrict__ be,
    const float* __restrict__ Wm,
    float* __restrict__ scoreS, float* __restrict__ scoreE,
    int* __restrict__ flags)
{
    __shared__ float Wp[(H_ / 2) * 8];   // 16 KB: W[k][n] packed pairwise in k
    __shared__ float At[WAVES * ATILE];  // 17 KB: per-wave A staging / D extract

    const int tid  = threadIdx.x;
    const int lane = tid & 31;
    const int wave = tid >> 5;

    // Pack weights: Wp[(k>>1)*8 + n*2 + (k&1)] = W[k][n], n: 0,1 -> Ws, 2,3 -> We
    for (int idx = tid; idx < H_ * 4; idx += 128) {
        int k = idx >> 2, n = idx & 3;
        float v = (n < 2) ? Ws[k * 2 + n] : We[k * 2 + (n - 2)];
        Wp[((k >> 1) << 3) + (n << 1) + (k & 1)] = v;
    }
    __syncthreads();

    const int tokBase = blockIdx.x * (WAVES * TPW) + wave * TPW;
    float* at = &At[wave * ATILE];

    // A fragment: lanes 0-15 rows M=lane, K pair {0,1}; lanes 16-31 K pair {2,3}
    const int  m    = lane & 15;
    const int  koff = (lane >> 4) << 1;       // 0 or 2
    // B fragment: N = lane&15 (cols 0..3 live, rest zero), same K-pair split
    const int  n    = lane & 15;
    const bool nact = (n < 4);
    const int  nn   = nact ? n : 0;

    v8f acc = {0.f, 0.f, 0.f, 0.f, 0.f, 0.f, 0.f, 0.f};

    for (int kc = 0; kc < H_; kc += KC) {
        // Coalesced staging: 16 tokens x 64 floats = 256 float4
        #pragma unroll
        for (int q = 0; q < 8; ++q) {
            int t4 = q * 32 + lane;
            int row = t4 >> 4, c4 = t4 & 15;
            v4f d = *(const v4f*)(rep + (size_t)(tokBase + row) * H_ + kc + c4 * 4);
            *(v4f*)(at + row * AROW + c4 * 4) = d;
        }
        // 16 x (K=4) WMMA steps over the chunk
        #pragma unroll
        for (int k = 0; k < KC / 4; ++k) {
            v2f a = *(const v2f*)(at + m * AROW + k * 4 + koff);
            int kk = kc + k * 4 + koff;                       // even
            v2f bw = *(const v2f*)(Wp + ((kk >> 1) << 3) + (nn << 1));
            v2f bf;
            bf.x = nact ? bw.x : 0.f;                          // v_cndmask, no EXEC change
            bf.y = nact ? bw.y : 0.f;
            acc = __builtin_amdgcn_wmma_f32_16x16x4_f32(
                false, a, false, bf, (short)0, acc, false, false);
        }
    }

    // D tile (16x16 f32): lanes 0-15 -> M=v, N=lane; lanes 16-31 -> M=v+8.
    // Spill through LDS (same-wave DS ops are in-order; no barrier needed).
    const int col = lane & 15;
    const int rb  = (lane >> 4) * 8;
    #pragma unroll
    for (int v = 0; v < 8; ++v) at[(rb + v) * 17 + col] = acc[v];

    if (lane < 16) {
        const int t = tokBase + lane;
        float sl0 = at[lane * 17 + 0] + bs[0];
        float sl1 = at[lane * 17 + 1] + bs[1];
        float el0 = at[lane * 17 + 2] + be[0];
        float el1 = at[lane * 17 + 3] + be[1];
        int mk = mask[t];
        int f = 0;
        if (mk != 0 && sl0 <= sl1) f |= 1;   // start candidate
        if (mk != 0 && el0 <= el1) f |= 2;   // end candidate
        scoreS[t] = sl0 * Wm[0] + sl1 * Wm[1];
        scoreE[t] = el0 * Wm[2] + el1 * Wm[3];
        flags[t]  = f;
    }
}

// ---------------- Phase 2: pairwise scores, store-bandwidth bound ----------
__global__ __launch_bounds__(256) void phase2_pairs(
    const float* __restrict__ scoreS, const float* __restrict__ scoreE,
    const int* __restrict__ flags, const float* __restrict__ bm,
    float* __restrict__ outValid, float* __restrict__ outScore)
{
    const int jb = (blockIdx.x * 256 + threadIdx.x) * 4;  // 4 j's per thread
    const int i  = blockIdx.y;
    const int b  = blockIdx.z;
    const int tb = b * S_;

    const float ss  = scoreS[tb + i];
    const bool  si  = (flags[tb + i] & 1) != 0;
    const float bmv = bm[0];

    v4f se = *(const v4f*)(scoreE + tb + jb);
    v4i fj = *(const v4i*)(flags + tb + jb);

    v4f ov, os;
    #pragma unroll
    for (int c = 0; c < 4; ++c) {
        float p = ss + se[c] + bmv;
        bool v = si && ((fj[c] & 2) != 0) && ((jb + c) >= i) && (p > 0.f);
        ov[c] = v ? 1.f : 0.f;
        os[c] = v ? p : 0.f;
    }

    const size_t base = ((size_t)b * S_ + i) * S_ + jb;
    __builtin_nontemporal_store(ov, (v4f*)(outValid + base));
    __builtin_nontemporal_store(os, (v4f*)(outScore + base));
}

extern "C" void kernel_launch(void* const* d_in, const int* in_sizes, int n_in,
                              void* d_out, int out_size, void* d_ws, size_t ws_size,
                              hipStream_t stream) {
    const float* rep  = (const float*)d_in[0];
    const int*   mask = (const int*)  d_in[1];
    const float* Ws   = (const float*)d_in[2];
    const float* bs   = (const float*)d_in[3];
    const float* We   = (const float*)d_in[4];
    const float* be   = (const float*)d_in[5];
    const float* Wm   = (const float*)d_in[6];
    const float* bm   = (const float*)d_in[7];

    float* scoreS = (float*)d_ws;
    float* scoreE = scoreS + NTOK;
    int*   flags  = (int*)(scoreE + NTOK);

    float* outValid = (float*)d_out;
    float* outScore = outValid + (size_t)B_ * S_ * S_;

    phase1_wmma<<<NTOK / (WAVES * TPW), 128, 0, stream>>>(
        rep, mask, Ws, bs, We, be, Wm, scoreS, scoreE, flags);

    dim3 g2(S_ / (256 * 4), S_, B_);
    phase2_pairs<<<g2, 256, 0, stream>>>(
        scoreS, scoreE, flags, bm, outValid, outScore);
}